// TorsionLayer_53695681135139
// MI455X (gfx1250) — compile-verified
//
#include <hip/hip_runtime.h>
#include <hip/hip_bf16.h>

// TorsionLayer: reference ignores the input and broadcasts [1.0f, 2.0f] to
// [B, N, 2] (B=128, N=8192) -> 2,097,152 fp32 = 8 MiB of pure stores.
// Roofline floor on MI455X @ 23.3 TB/s: ~0.36 us. The [1,2] pattern tiles
// exactly into a float4 {1,2,1,2}: each lane emits 4 coalesced
// global_store_b128 (512 B per wave32 per instruction).
//
// Codegen refinements this round:
//  - blockDim hard-coded to 256 (compile-time) so the stride folds into the
//    24-bit instruction IOFFSET: all 4 stores share one index VGPR with
//    scale_offset + offset immediates; no 64-bit address-add chains, no
//    kernarg read of the workgroup size.
//  - constants stay as VALU inline 1.0/2.0 (no rodata).

struct __align__(16) f4 { float x, y, z, w; };

static constexpr int BLOCK = 256;   // 8 wave32s per block
static constexpr int UNROLL = 4;    // float4 stores per thread

static __device__ __forceinline__ f4 make_pattern() {
    f4 v;
    v.x = 1.0f;
    v.y = 2.0f;
    v.z = 1.0f;
    v.w = 2.0f;
    return v;
}

// Exact kernel: grid*BLOCK*UNROLL == n4. No guards.
__global__ void __launch_bounds__(BLOCK)
torsion_fill_exact(f4* __restrict__ out) {
    const f4 v = make_pattern();
    int base = blockIdx.x * (BLOCK * UNROLL) + threadIdx.x;
    out[base]             = v;   // scale_offset
    out[base + BLOCK]     = v;   // scale_offset offset:4096
    out[base + 2 * BLOCK] = v;   // scale_offset offset:8192
    out[base + 3 * BLOCK] = v;   // scale_offset offset:12288
}

// Guarded fallback for non-tiling sizes (never taken for the reference shape).
__global__ void __launch_bounds__(BLOCK)
torsion_fill_guarded(f4* __restrict__ out, int n4) {
    const f4 v = make_pattern();
    int base = blockIdx.x * (BLOCK * UNROLL) + threadIdx.x;
#pragma unroll
    for (int i = 0; i < UNROLL; ++i) {
        int idx = base + i * BLOCK;
        if (idx < n4) out[idx] = v;
    }
}

extern "C" void kernel_launch(void* const* d_in, const int* in_sizes, int n_in,
                              void* d_out, int out_size, void* d_ws, size_t ws_size,
                              hipStream_t stream) {
    (void)d_in; (void)in_sizes; (void)n_in; (void)d_ws; (void)ws_size;

    // out_size = B * N * 2 = 2,097,152 fp32 for the reference shape.
    int n4 = out_size >> 2;                    // float4 count (524,288)
    const int per_block = BLOCK * UNROLL;      // 1024 float4 per block

    if (n4 % per_block == 0 && n4 > 0) {
        int grid = n4 / per_block;             // = 512 for the reference shape
        torsion_fill_exact<<<grid, BLOCK, 0, stream>>>((f4*)d_out);
    } else {
        int grid = (n4 + per_block - 1) / per_block;
        if (grid < 1) grid = 1;
        torsion_fill_guarded<<<grid, BLOCK, 0, stream>>>((f4*)d_out, n4);
    }
}